// AttGRU_18073222381850
// MI455X (gfx1250) — compile-verified
//
#include <hip/hip_runtime.h>
#include <math.h>

// Problem constants (from reference): B=512, S=512, H=100
#define BSZ   512
#define SSZ   512
#define HDIM  100
#define HP    112          // H padded to 7*16 WMMA N-tiles
#define HS    114          // LDS row stride (even for b64 align, avoids bank conflicts)
#define NKC   25           // K chunks of 4 (100 = 25*4)
#define NJT   7            // N tiles of 16 (112 = 7*16)
#define MSZ   (NKC*NJT*64) // floats per weight matrix in WMMA-B layout = 11200
#define TPB   224          // 7 waves, one per N-tile
#define NPF   8            // prefetch regs per thread: ceil(16*100/224)

typedef __attribute__((ext_vector_type(2))) float v2f;
typedef __attribute__((ext_vector_type(8))) float v8f;

__device__ __forceinline__ float fast_sigmoid(float x) {
    return __builtin_amdgcn_rcpf(1.f + __expf(-x));
}
__device__ __forceinline__ float fast_tanh(float x) {
    // 1 - 2/(e^{2x}+1): exact limits at +-inf (e^{2x} -> inf / 0)
    return 1.f - 2.f * __builtin_amdgcn_rcpf(1.f + __expf(2.f * x));
}

// ---------------------------------------------------------------------------
// Kernel 1: attention scores + softmax.  One block per batch row.
// ---------------------------------------------------------------------------
__global__ __launch_bounds__(256)
void attn_softmax_kernel(const float* __restrict__ q,
                         const float* __restrict__ ctx,
                         float* __restrict__ scores)
{
    const int b   = blockIdx.x;
    const int tid = threadIdx.x;

    __shared__ float qs[HDIM];
    __shared__ float red[256];

    if (tid < HDIM) qs[tid] = q[(size_t)b * HDIM + tid];
    __syncthreads();

    float v0 = 0.f, v1 = 0.f;
    {
        const float* c0 = ctx + ((size_t)b * SSZ + tid) * HDIM;
        const float* c1 = ctx + ((size_t)b * SSZ + tid + 256) * HDIM;
        #pragma unroll 4
        for (int k = 0; k < HDIM; ++k) {
            v0 += qs[k] * c0[k];
            v1 += qs[k] * c1[k];
        }
    }

    red[tid] = fmaxf(v0, v1);
    __syncthreads();
    for (int off = 128; off > 0; off >>= 1) {
        if (tid < off) red[tid] = fmaxf(red[tid], red[tid + off]);
        __syncthreads();
    }
    const float mx = red[0];
    __syncthreads();

    const float e0 = expf(v0 - mx);
    const float e1 = expf(v1 - mx);

    red[tid] = e0 + e1;
    __syncthreads();
    for (int off = 128; off > 0; off >>= 1) {
        if (tid < off) red[tid] += red[tid + off];
        __syncthreads();
    }
    const float inv = 1.f / red[0];

    scores[(size_t)b * SSZ + tid]       = e0 * inv;
    scores[(size_t)b * SSZ + tid + 256] = e1 * inv;
}

// ---------------------------------------------------------------------------
// Kernel 2: fused attention-GRU scan.  32 blocks x 16 batch rows, 7 waves,
// one WMMA N-tile per wave.  Four independent 25-deep WMMA accumulation
// chains per wave; double-buffered register prefetch of the next step's
// context tile; per-lane h slice kept in registers.
// ---------------------------------------------------------------------------
__global__ __launch_bounds__(TPB)
void gru_scan_kernel(const float* __restrict__ ctx,
                     const float* __restrict__ h0,
                     const float* __restrict__ Wr,
                     const float* __restrict__ Ur,
                     const float* __restrict__ W,
                     const float* __restrict__ U,
                     const float* __restrict__ bWr,
                     const float* __restrict__ bUr,
                     const float* __restrict__ bW,
                     const float* __restrict__ bU,
                     const float* __restrict__ scores,
                     float* __restrict__ out)
{
    __shared__ __align__(16) float matB[4 * MSZ];      // Wr,Ur,W,U in WMMA-B layout
    __shared__ __align__(16) float h_lds[16 * HS];     // hidden-state tile
    __shared__ __align__(16) float c_buf[2][16 * HS];  // double-buffered context tile
    __shared__ float biasR[HP];
    __shared__ float biasW[HP];
    __shared__ float biasU[HP];
    __shared__ float g_lds[16];

    const int tid = threadIdx.x;
    const int b0  = blockIdx.x * 16;

    // ---- init: swizzle weights into WMMA-B layout ----
    for (int e = tid; e < 4 * MSZ; e += TPB) {
        const int mi  = e / MSZ;
        const int rem = e % MSZ;
        const int kk  = rem / (NJT * 64);
        const int r2  = rem % (NJT * 64);
        const int j   = r2 >> 6;
        const int o   = r2 & 63;
        const int k4  = ((o >> 5) << 1) | (o & 1);
        const int n16 = (o & 31) >> 1;
        const int n   = j * 16 + n16;
        const int k   = kk * 4 + k4;
        const float* mp = (mi == 0) ? Wr : (mi == 1) ? Ur : (mi == 2) ? W : U;
        matB[e] = (n < HDIM) ? mp[(size_t)n * HDIM + k] : 0.f;   // B[k][n] = Mat[n][k]
    }
    for (int n = tid; n < HP; n += TPB) {
        biasR[n] = (n < HDIM) ? (bWr[n] + bUr[n]) : 0.f;
        biasW[n] = (n < HDIM) ? bW[n] : 0.f;
        biasU[n] = (n < HDIM) ? bU[n] : 0.f;
    }
    for (int e = tid; e < 16 * HP; e += TPB) {
        const int m = e / HP, n = e % HP;
        h_lds[m * HS + n] = (n < HDIM) ? h0[(size_t)(b0 + m) * HDIM + n] : 0.f;
    }
    __syncthreads();

    const int lane  = tid & 31;
    const int w     = tid >> 5;          // this wave's N-tile (0..6)
    const int nlo   = lane & 15;
    const int khalf = (lane >> 4) << 1;  // 0 or 2
    const int mbase = (lane >> 4) << 3;  // C/D: lanes 16-31 hold M+8
    const int col   = w * 16 + nlo;

    const float bR  = biasR[col];
    const float bWc = biasW[col];
    const float bUc = biasU[col];

    // per-lane slice of h (exactly the elements this lane computes/blends)
    float hreg[8];
    #pragma unroll
    for (int v = 0; v < 8; ++v)
        hreg[v] = h_lds[(mbase + v) * HS + col];

    // ---- staging descriptors (16x100 tile split over 224 threads) ----
    int    loff_[NPF];
    size_t goff_[NPF];
    bool   val_[NPF];
    #pragma unroll
    for (int i = 0; i < NPF; ++i) {
        const int e = tid + i * TPB;
        val_[i] = (e < 16 * HDIM);
        const int m = val_[i] ? (e / HDIM) : 0;
        const int n = val_[i] ? (e % HDIM) : 0;
        loff_[i] = m * HS + n;
        goff_[i] = (size_t)(b0 + m) * SSZ * HDIM + n;
    }

    // prefetch step 0 into registers
    float cpf[NPF];
    #pragma unroll
    for (int i = 0; i < NPF; ++i)
        if (val_[i]) cpf[i] = ctx[goff_[i]];
    float gpf = (tid < 16) ? scores[(size_t)(b0 + tid) * SSZ] : 0.f;

    for (int s = 0; s < SSZ; ++s) {
        // commit prefetched tile to this step's LDS buffer
        float* cb = &c_buf[s & 1][0];
        #pragma unroll
        for (int i = 0; i < NPF; ++i)
            if (val_[i]) cb[loff_[i]] = cpf[i];
        if (tid < 16) g_lds[tid] = gpf;

        // launch prefetch of the NEXT step's tile (overlaps the WMMA work)
        if (s + 1 < SSZ) {
            #pragma unroll
            for (int i = 0; i < NPF; ++i)
                if (val_[i]) cpf[i] = ctx[goff_[i] + (size_t)(s + 1) * HDIM];
            if (tid < 16) gpf = scores[(size_t)(b0 + tid) * SSZ + s + 1];
        }
        __syncthreads();

        // four independent accumulation chains (depth 25 each)
        v8f accRc = {0.f, 0.f, 0.f, 0.f, 0.f, 0.f, 0.f, 0.f};
        v8f accRh = {0.f, 0.f, 0.f, 0.f, 0.f, 0.f, 0.f, 0.f};
        v8f accCW = {0.f, 0.f, 0.f, 0.f, 0.f, 0.f, 0.f, 0.f};
        v8f accHU = {0.f, 0.f, 0.f, 0.f, 0.f, 0.f, 0.f, 0.f};

        #pragma unroll
        for (int kk = 0; kk < NKC; ++kk) {
            const int aoff = nlo * HS + kk * 4 + khalf;
            const v2f a_c = *(const v2f*)&cb[aoff];
            const v2f a_h = *(const v2f*)&h_lds[aoff];

            const int boff = (kk * NJT + w) * 64 + (khalf << 4) + nlo * 2;
            const v2f b_wr = *(const v2f*)&matB[0 * MSZ + boff];
            const v2f b_ur = *(const v2f*)&matB[1 * MSZ + boff];
            const v2f b_w  = *(const v2f*)&matB[2 * MSZ + boff];
            const v2f b_u  = *(const v2f*)&matB[3 * MSZ + boff];

            accRc = __builtin_amdgcn_wmma_f32_16x16x4_f32(false, a_c, false, b_wr,
                                                          (short)0, accRc, false, false);
            accRh = __builtin_amdgcn_wmma_f32_16x16x4_f32(false, a_h, false, b_ur,
                                                          (short)0, accRh, false, false);
            accCW = __builtin_amdgcn_wmma_f32_16x16x4_f32(false, a_c, false, b_w,
                                                          (short)0, accCW, false, false);
            accHU = __builtin_amdgcn_wmma_f32_16x16x4_f32(false, a_h, false, b_u,
                                                          (short)0, accHU, false, false);
        }
        __syncthreads();   // all h/c LDS reads done before h is overwritten

        #pragma unroll
        for (int v = 0; v < 8; ++v) {
            const int m    = mbase + v;
            const float r  = fast_sigmoid(accRc[v] + accRh[v] + bR);
            const float ht = fast_tanh(accCW[v] + bWc + r * (accHU[v] + bUc));
            const float g  = g_lds[m];
            hreg[v] = g * ht + (1.f - g) * hreg[v];
        }
        #pragma unroll
        for (int v = 0; v < 8; ++v)
            h_lds[(mbase + v) * HS + col] = hreg[v];   // each lane owns disjoint (m,col)
        // next iteration's post-staging __syncthreads orders these writes vs. reads
    }

    __syncthreads();
    for (int e = tid; e < 16 * HDIM; e += TPB) {
        const int m = e / HDIM, n = e % HDIM;
        out[(size_t)(b0 + m) * HDIM + n] = h_lds[m * HS + n];
    }
}

// ---------------------------------------------------------------------------
// Host launcher.  Inputs in setup_inputs() order.
// d_out = [output (B*H floats)] ++ [scores (B*S floats)]
// ---------------------------------------------------------------------------
extern "C" void kernel_launch(void* const* d_in, const int* in_sizes, int n_in,
                              void* d_out, int out_size, void* d_ws, size_t ws_size,
                              hipStream_t stream)
{
    const float* query       = (const float*)d_in[0];
    const float* context     = (const float*)d_in[1];
    const float* init_hidden = (const float*)d_in[2];
    const float* Wr          = (const float*)d_in[3];
    const float* bWr         = (const float*)d_in[4];
    const float* Ur          = (const float*)d_in[5];
    const float* bUr         = (const float*)d_in[6];
    const float* W           = (const float*)d_in[7];
    const float* bW          = (const float*)d_in[8];
    const float* U           = (const float*)d_in[9];
    const float* bU          = (const float*)d_in[10];

    float* out    = (float*)d_out;
    float* scores = out + (size_t)BSZ * HDIM;   // second tuple element

    attn_softmax_kernel<<<BSZ, 256, 0, stream>>>(query, context, scores);

    gru_scan_kernel<<<BSZ / 16, TPB, 0, stream>>>(
        context, init_hidden, Wr, Ur, W, U, bWr, bUr, bW, bU, scores, out);
}